// FullTGCRN_79912161509734
// MI455X (gfx1250) — compile-verified
//
#include <hip/hip_runtime.h>
#include <cstdint>

#define DEVFN __device__ __forceinline__

typedef float v2f __attribute__((ext_vector_type(2)));
typedef float v8f __attribute__((ext_vector_type(8)));

// ---- problem constants ----
constexpr int B_ = 8, T_ = 12, N_ = 2000, H_ = 64, E_ = 10, CIN = 65, H2 = 128;
constexpr int M_FF = B_ * T_ * N_;            // 192000 rows for the 64x64 FF GEMMs

// ---- workspace layout (float elements). Total ~58.6M floats (~234 MB). ----
constexpr size_t A_OFF    = 0;                      constexpr size_t A_SZ   = (size_t)N_ * N_;
constexpr size_t BG_OFF   = A_OFF + A_SZ;           constexpr size_t BG_SZ  = (size_t)N_ * H2;
constexpr size_t BU_OFF   = BG_OFF + BG_SZ;         constexpr size_t BU_SZ  = (size_t)N_ * H_;
constexpr size_t ST_OFF   = BU_OFF + BU_SZ;         constexpr size_t ST_SZ  = (size_t)B_ * N_ * H_;
constexpr size_t SEQ_OFF  = ST_OFF + ST_SZ;         constexpr size_t SEQ_SZ = (size_t)B_ * T_ * N_ * H_;
constexpr size_t S_OFF    = SEQ_OFF + SEQ_SZ;       constexpr size_t S_SZ   = 256;
constexpr size_t AD_OFF   = S_OFF + S_SZ;           constexpr size_t AD_SZ  = A_SZ;   // A .* drop scratch
constexpr size_t R_OFF    = AD_OFF + AD_SZ;
// GRU-phase scratch (aliased later by V1/V2/V3)
constexpr size_t INS_OFF  = R_OFF;                  constexpr size_t INS_SZ  = (size_t)B_ * N_ * CIN;
constexpr size_t CAND_OFF = INS_OFF + INS_SZ;       constexpr size_t CAND_SZ = INS_SZ;
constexpr size_t AX_OFF   = CAND_OFF + CAND_SZ;     constexpr size_t AX_SZ   = (size_t)B_ * N_ * H2;
constexpr size_t XG_OFF   = AX_OFF + AX_SZ;         constexpr size_t XG_SZ   = (size_t)B_ * N_ * 256;
constexpr size_t H0G_OFF  = XG_OFF + XG_SZ;         constexpr size_t H0G_SZ  = (size_t)B_ * N_ * H2;
constexpr size_t ZR_OFF   = H0G_OFF + H0G_SZ;       constexpr size_t ZR_SZ   = H0G_SZ;
constexpr size_t H0U_OFF  = ZR_OFF + ZR_SZ;         constexpr size_t H0U_SZ  = (size_t)B_ * N_ * H_;
constexpr size_t HU_OFF   = H0U_OFF + H0U_SZ;
// temporal-phase buffers alias the GRU scratch region
constexpr size_t V1_OFF = R_OFF;
constexpr size_t V2_OFF = V1_OFF + SEQ_SZ;
constexpr size_t V3_OFF = V2_OFF + SEQ_SZ;

// ---- deterministic dropout hash (replaces jax threefry; deterministic per call) ----
DEVFN uint32_t hash3(uint32_t a, uint32_t b, uint32_t c) {
  uint32_t x = a * 0x9E3779B1u ^ (b * 0x85EBCA77u + 0x165667B1u) ^ (c * 0xC2B2AE3Du + 0x27D4EB2Fu);
  x ^= x >> 15; x *= 0x2C1B3C6Du; x ^= x >> 12; x *= 0x297A2D39u; x ^= x >> 15;
  return x;
}
DEVFN float dropmul(int tag, int i, int j) {
  uint32_t h = hash3((uint32_t)(tag + 1), (uint32_t)i, (uint32_t)j);
  float u = (float)(h >> 8) * (1.0f / 16777216.0f);
  return fmaxf(u - 0.4f, 0.0f);   // relu(uniform - DROPRATE)
}

// ---- fp32 WMMA 16x16x4 (D = A(16x4) x B(4x16) + C) ----
DEVFN v8f wmma_f32(v2f a, v2f b, v8f c) {
  return __builtin_amdgcn_wmma_f32_16x16x4_f32(false, a, false, b, (short)0, c, false, false);
}

// =============== adjacency: A = softmax(relu(ne @ ne^T), axis=1) ===============
__global__ __launch_bounds__(256) void adj_kernel(const float* __restrict__ ne, float* __restrict__ A) {
  __shared__ float nei[16];
  __shared__ float red[256];
  const int i = blockIdx.x;
  const int tid = threadIdx.x;
  if (tid < E_) nei[tid] = ne[(size_t)i * E_ + tid];
  __syncthreads();
  float lmax = 0.f;
  for (int j = tid; j < N_; j += 256) {
    const float* nj = ne + (size_t)j * E_;
    float s = 0.f;
#pragma unroll
    for (int d = 0; d < E_; ++d) s += nei[d] * nj[d];
    s = fmaxf(s, 0.f);
    A[(size_t)i * N_ + j] = s;
    lmax = fmaxf(lmax, s);
  }
  red[tid] = lmax; __syncthreads();
  for (int o = 128; o > 0; o >>= 1) { if (tid < o) red[tid] = fmaxf(red[tid], red[tid + o]); __syncthreads(); }
  const float m = red[0];
  __syncthreads();
  float lsum = 0.f;
  for (int j = tid; j < N_; j += 256) lsum += __expf(A[(size_t)i * N_ + j] - m);
  red[tid] = lsum; __syncthreads();
  for (int o = 128; o > 0; o >>= 1) { if (tid < o) red[tid] += red[tid + o]; __syncthreads(); }
  const float inv = 1.f / red[0];
  for (int j = tid; j < N_; j += 256) A[(size_t)i * N_ + j] = __expf(A[(size_t)i * N_ + j] - m) * inv;
}

// =============== Adrop = A .* drop(tag) (hash hoisted out of the GEMM hot loop) ===============
__global__ void drop_adj_kernel(const float* __restrict__ A, float* __restrict__ Ad, int tag) {
  long long idx = (long long)blockIdx.x * 256 + threadIdx.x;
  if (idx >= (long long)N_ * N_) return;
  int i = (int)(idx / N_), j = (int)(idx % N_);
  Ad[idx] = A[idx] * dropmul(tag, i, j);
}

// =============== bias[n,o] = sum_d ne[n,d] * bp[d,o] ===============
__global__ void bias_kernel(const float* __restrict__ ne, const float* __restrict__ bp,
                            float* __restrict__ out, int O) {
  long long idx = (long long)blockIdx.x * 256 + threadIdx.x;
  if (idx >= (long long)N_ * O) return;
  int o = (int)(idx % O); int n = (int)(idx / O);
  float s = 0.f;
#pragma unroll
  for (int d = 0; d < E_; ++d) s += ne[(size_t)n * E_ + d] * bp[(size_t)d * O + o];
  out[idx] = s;
}

// =============== temporal adjacency S (12x12): softmax(relu(te te^T)) + I, tril ===============
__global__ void timeS_kernel(const float* __restrict__ te, float* __restrict__ S) {
  int t = threadIdx.x;
  if (t >= T_) return;
  float row[T_];
  float mx = 0.f;
  for (int j = 0; j < T_; ++j) {
    float s = 0.f;
    for (int d = 0; d < E_; ++d) s += te[t * E_ + d] * te[j * E_ + d];
    s = fmaxf(s, 0.f);
    row[j] = s; mx = fmaxf(mx, s);
  }
  float sum = 0.f;
  for (int j = 0; j < T_; ++j) { row[j] = __expf(row[j] - mx); sum += row[j]; }
  float inv = 1.f / sum;
  for (int j = 0; j < T_; ++j) {
    float v = row[j] * inv;
    if (j == t) v += 1.f;
    if (j > t)  v  = 0.f;
    S[t * T_ + j] = v;
  }
}

__global__ void copy_kernel(float* __restrict__ dst, const float* __restrict__ src, long long count) {
  long long idx = (long long)blockIdx.x * 256 + threadIdx.x;
  if (idx < count) dst[idx] = src[idx];
}

// =============== pack concat([x_t, (zr?)*state]) -> (B,N,CIN) ===============
__global__ void pack_in_kernel(const float* __restrict__ x, int t, const float* __restrict__ state,
                               const float* __restrict__ zr, float* __restrict__ out) {
  long long idx = (long long)blockIdx.x * 256 + threadIdx.x;
  if (idx >= (long long)B_ * N_ * CIN) return;
  int c = (int)(idx % CIN);
  long long bn = idx / CIN;
  int n = (int)(bn % N_);
  int b = (int)(bn / N_);
  float v;
  if (c == 0) {
    v = x[((size_t)b * T_ + t) * N_ + n];                   // F_IN == 1
  } else {
    float st = state[((size_t)b * N_ + n) * H_ + (c - 1)];
    if (zr) st *= zr[((size_t)b * N_ + n) * H2 + (c - 1)];  // z gate (first H of zr)
    v = st;
  }
  out[idx] = v;
}

// =============== xg = [sup0 * X0 | X1] -> (B,N,2C); sup0 = I (or diag(drop)) ===============
__global__ void pack_xg_kernel(const float* __restrict__ X0, const float* __restrict__ X1,
                               float* __restrict__ out, int C, int dropTag) {
  long long idx = (long long)blockIdx.x * 256 + threadIdx.x;
  if (idx >= (long long)B_ * N_ * 2 * C) return;
  int j = (int)(idx % (2 * C));
  long long bn = idx / (2 * C);
  int n = (int)(bn % N_);
  int b = (int)(bn / N_);
  float v;
  if (j < C) {
    v = X0[((size_t)b * N_ + n) * C + j];
    if (dropTag >= 0) v *= dropmul(dropTag, n, n);          // I .* drop -> diag
  } else {
    v = X1[((size_t)b * N_ + n) * C + (j - C)];
  }
  out[idx] = v;
}

// =============== graph GEMM: Y[b] = A @ X[b]; 64x64 block tile, fp32 WMMA ===============
// 8 waves: wm=wave>>1 (4 M-subtiles), wn=wave&1; each wave owns 1 M x 2 N 16x16 tiles
// so the A fragment is reused across two v_wmma per k-step.
__global__ __launch_bounds__(256) void graph_gemm(const float* __restrict__ A, const float* __restrict__ X,
                                                  float* __restrict__ Y, int C) {
  __shared__ float As[64][20];
  __shared__ float Bs[16][72];
  const int tid = threadIdx.x;
  const int lane = tid & 31, wave = tid >> 5;
  const int wm = wave >> 1, wn = wave & 1;
  const int rowBase = blockIdx.x * 64;
  const int colBase = blockIdx.y * 64;
  const int b = blockIdx.z;
  const int mrow = lane & 15;
  const int khalf = (lane >> 4) << 1;
  const bool cvec = ((C & 3) == 0);              // fp32x4 path only when rows are 16B aligned
  const v8f vz = {0.f, 0.f, 0.f, 0.f, 0.f, 0.f, 0.f, 0.f};
  v8f acc0 = vz, acc1 = vz;
  for (int kk = 0; kk < N_; kk += 16) {
    { // A tile 64x16, one float4 per thread -> global_load_b128
      int r = tid >> 2, c4 = (tid & 3) * 4;
      int gr = rowBase + r;
      float4 v = make_float4(0.f, 0.f, 0.f, 0.f);
      if (gr < N_) v = *(const float4*)&A[(size_t)gr * N_ + kk + c4];
      As[r][c4 + 0] = v.x; As[r][c4 + 1] = v.y; As[r][c4 + 2] = v.z; As[r][c4 + 3] = v.w;
    }
    { // X tile 16x64, one float4 per thread where aligned
      int r = tid >> 4, c4 = (tid & 15) * 4;
      int gc = colBase + c4;
      const float* xrow = &X[((size_t)b * N_ + kk + r) * C];
      float4 v;
      if (cvec && gc + 3 < C) {
        v = *(const float4*)&xrow[gc];
      } else {
        v.x = (gc + 0 < C) ? xrow[gc + 0] : 0.f;
        v.y = (gc + 1 < C) ? xrow[gc + 1] : 0.f;
        v.z = (gc + 2 < C) ? xrow[gc + 2] : 0.f;
        v.w = (gc + 3 < C) ? xrow[gc + 3] : 0.f;
      }
      Bs[r][c4 + 0] = v.x; Bs[r][c4 + 1] = v.y; Bs[r][c4 + 2] = v.z; Bs[r][c4 + 3] = v.w;
    }
    if (kk + 16 < N_) {                          // CDNA5 global_prefetch_b8 for next A tile
      int pr = rowBase + (tid >> 2);
      if (pr < N_) __builtin_prefetch(&A[(size_t)pr * N_ + kk + 16], 0, 1);
    }
    __syncthreads();
#pragma unroll
    for (int ks = 0; ks < 16; ks += 4) {
      v2f a, b0, b1;
      a.x  = As[wm * 16 + mrow][ks + khalf];
      a.y  = As[wm * 16 + mrow][ks + khalf + 1];
      b0.x = Bs[ks + khalf][wn * 32 + mrow];
      b0.y = Bs[ks + khalf + 1][wn * 32 + mrow];
      b1.x = Bs[ks + khalf][wn * 32 + 16 + mrow];
      b1.y = Bs[ks + khalf + 1][wn * 32 + 16 + mrow];
      acc0 = wmma_f32(a, b0, acc0);
      acc1 = wmma_f32(a, b1, acc1);
    }
    __syncthreads();
  }
  const int mo = (lane >> 4) * 8;
#pragma unroll
  for (int r = 0; r < 8; ++r) {
    int gr  = rowBase + wm * 16 + mo + r;
    int gc0 = colBase + wn * 32 + mrow;
    int gc1 = gc0 + 16;
    if (gr < N_) {
      if (gc0 < C) Y[((size_t)b * N_ + gr) * C + gc0] = acc0[r];
      if (gc1 < C) Y[((size_t)b * N_ + gr) * C + gc1] = acc1[r];
    }
  }
}

// =============== per-node GEMM: h[b,n,:] = xg[b,n,:] @ (sum_d ne[n,d] wp[d]) + bias (+add, act) ===============
__global__ __launch_bounds__(256) void node_gemm(const float* __restrict__ xg, const float* __restrict__ wp,
                                                 const float* __restrict__ ne, const float* __restrict__ bias,
                                                 const float* __restrict__ addsrc, float* __restrict__ out,
                                                 int J, int C, int O, int act) {
  __shared__ float Wt[16][132];
  __shared__ float Xs[16][20];
  __shared__ float nesh[16];
  const int n = blockIdx.x;
  const int tid = threadIdx.x;
  const int lane = tid & 31, wave = tid >> 5;
  if (tid < E_) nesh[tid] = ne[(size_t)n * E_ + tid];
  __syncthreads();
  const bool active = (wave * 16 < O);           // wave owns one 16-wide output tile
  const int mrow = lane & 15;
  const int khalf = (lane >> 4) << 1;
  v8f acc = {0.f, 0.f, 0.f, 0.f, 0.f, 0.f, 0.f, 0.f};
  const int ksteps = (J + 15) / 16;
  const size_t dstride = (size_t)2 * C * O;      // wp[d,k,i,o] stride over d
  for (int jt = 0; jt < ksteps; ++jt) {
    const int j0 = jt * 16;
    for (int idx = tid; idx < 16 * O; idx += 256) {   // blend W tile into LDS
      int r = idx / O, c = idx - r * O;
      int j = j0 + r;
      float w = 0.f;
      if (j < J) {
        int k = j / C, i = j - k * C;
        size_t base = ((size_t)k * C + i) * O + c;
#pragma unroll
        for (int d = 0; d < E_; ++d) w += nesh[d] * wp[base + (size_t)d * dstride];
      }
      Wt[r][c] = w;
    }
    {
      int r = tid >> 4, c = tid & 15;            // 16x16 activation tile (rows 8..15 zero)
      int j = j0 + c;
      Xs[r][c] = (r < B_ && j < J) ? xg[((size_t)r * N_ + n) * J + j] : 0.f;
    }
    __syncthreads();
    if (active) {
#pragma unroll
      for (int ks = 0; ks < 16; ks += 4) {
        v2f a, bv;
        a.x = Xs[mrow][ks + khalf];
        a.y = Xs[mrow][ks + khalf + 1];
        bv.x = Wt[ks + khalf][wave * 16 + mrow];
        bv.y = Wt[ks + khalf + 1][wave * 16 + mrow];
        acc = wmma_f32(a, bv, acc);
      }
    }
    __syncthreads();
  }
  if (active) {
    const int mo = (lane >> 4) * 8;
#pragma unroll
    for (int r = 0; r < 8; ++r) {
      int bb = mo + r;
      if (bb < B_) {
        int o = wave * 16 + mrow;
        float v = acc[r] + bias[(size_t)n * O + o];
        if (addsrc) v += addsrc[((size_t)bb * N_ + n) * O + o];
        if (act == 1) v = 1.f / (1.f + __expf(-v));
        else if (act == 2) v = tanhf(v);
        out[((size_t)bb * N_ + n) * O + o] = v;
      }
    }
  }
}

// =============== GRU combine: state = r*state + (1-r)*hc; seq[:,t] = state ===============
__global__ void gru_update_kernel(const float* __restrict__ zr, const float* __restrict__ hu,
                                  float* __restrict__ state, float* __restrict__ seq, int t) {
  long long idx = (long long)blockIdx.x * 256 + threadIdx.x;
  if (idx >= (long long)B_ * N_ * H_) return;
  int h = (int)(idx % H_);
  int n = (int)((idx / H_) % N_);
  int b = (int)(idx / ((long long)N_ * H_));
  float r = zr[((size_t)b * N_ + n) * H2 + H_ + h];
  float ns = r * state[idx] + (1.f - r) * hu[idx];
  state[idx] = ns;
  seq[(((size_t)b * T_ + t) * N_ + n) * H_ + h] = ns;
}

// =============== M x 64 @ 64 x 64 + bias (opt. relu), fp32 WMMA ===============
__global__ __launch_bounds__(256) void gemm_ff(const float* __restrict__ X, const float* __restrict__ W,
                                               const float* __restrict__ bias, float* __restrict__ Y,
                                               int M, int act) {
  __shared__ float Ws[64][68];
  __shared__ float Xs[128][68];
  const int tid = threadIdx.x;
  const int lane = tid & 31, wave = tid >> 5;
  const int mBase = blockIdx.x * 128;
  for (int v4 = tid; v4 < 1024; v4 += 256) {     // W: 64x64 as float4 -> global_load_b128
    int r = v4 >> 4, c4 = (v4 & 15) * 4;
    float4 w = *(const float4*)&W[(size_t)r * 64 + c4];
    Ws[r][c4 + 0] = w.x; Ws[r][c4 + 1] = w.y; Ws[r][c4 + 2] = w.z; Ws[r][c4 + 3] = w.w;
  }
  for (int v4 = tid; v4 < 2048; v4 += 256) {     // X: 128x64 as float4
    int r = v4 >> 4, c4 = (v4 & 15) * 4;
    int gr = mBase + r;
    float4 v = make_float4(0.f, 0.f, 0.f, 0.f);
    if (gr < M) v = *(const float4*)&X[(size_t)gr * 64 + c4];
    Xs[r][c4 + 0] = v.x; Xs[r][c4 + 1] = v.y; Xs[r][c4 + 2] = v.z; Xs[r][c4 + 3] = v.w;
  }
  __syncthreads();
  const int mrow = lane & 15;
  const int khalf = (lane >> 4) << 1;
  const v8f vz = {0.f, 0.f, 0.f, 0.f, 0.f, 0.f, 0.f, 0.f};
  v8f acc[4];
#pragma unroll
  for (int ct = 0; ct < 4; ++ct) acc[ct] = vz;
  for (int ks = 0; ks < 64; ks += 4) {
    v2f a;
    a.x = Xs[wave * 16 + mrow][ks + khalf];
    a.y = Xs[wave * 16 + mrow][ks + khalf + 1];
#pragma unroll
    for (int ct = 0; ct < 4; ++ct) {
      v2f bv;
      bv.x = Ws[ks + khalf][ct * 16 + mrow];
      bv.y = Ws[ks + khalf + 1][ct * 16 + mrow];
      acc[ct] = wmma_f32(a, bv, acc[ct]);
    }
  }
  const int mo = (lane >> 4) * 8;
#pragma unroll
  for (int ct = 0; ct < 4; ++ct) {
#pragma unroll
    for (int r = 0; r < 8; ++r) {
      int gr = mBase + wave * 16 + mo + r;
      int gc = ct * 16 + mrow;
      if (gr < M) {
        float v = acc[ct][r] + bias[gc];
        if (act == 1) v = fmaxf(v, 0.f);
        Y[(size_t)gr * 64 + gc] = v;
      }
    }
  }
}

// =============== temporal mix over T: out[b,t] = sum_s S[t,s](*drop) in[b,s] (+residual) ===============
__global__ void time_mix_kernel(const float* __restrict__ S, const float* __restrict__ in,
                                float* __restrict__ out, int mode) {
  __shared__ float Ssh[T_ * T_];
  if ((int)threadIdx.x < T_ * T_) Ssh[threadIdx.x] = S[threadIdx.x];
  __syncthreads();
  long long idx = (long long)blockIdx.x * 256 + threadIdx.x;
  if (idx >= (long long)B_ * T_ * N_ * H_) return;
  int d = (int)(idx % H_);
  int n = (int)((idx / H_) % N_);
  int t = (int)((idx / ((long long)N_ * H_)) % T_);
  int b = (int)(idx / ((long long)T_ * N_ * H_));
  float s = 0.f;
#pragma unroll
  for (int ss = 0; ss < T_; ++ss) {
    float w = Ssh[t * T_ + ss];
    if (mode) w *= dropmul(1000, t, ss);
    s += w * in[(((size_t)b * T_ + ss) * N_ + n) * H_ + d];
  }
  if (mode) s += in[idx];                        // + v0 residual
  out[idx] = s;
}

// =============== layernorm over H=64 (one wave per vector), optional residual add ===============
__global__ __launch_bounds__(256) void ln_kernel(const float* __restrict__ src, const float* __restrict__ addsrc,
                                                 const float* __restrict__ g, const float* __restrict__ bt,
                                                 float* __restrict__ out, long long nvec) {
  const int lane = threadIdx.x & 31;
  const int wave = threadIdx.x >> 5;
  long long vec = (long long)blockIdx.x * 8 + wave;
  if (vec >= nvec) return;
  size_t base = (size_t)vec * H_;
  float x0 = src[base + lane], x1 = src[base + lane + 32];
  if (addsrc) { x0 += addsrc[base + lane]; x1 += addsrc[base + lane + 32]; }
  float s = x0 + x1;
#pragma unroll
  for (int off = 16; off > 0; off >>= 1) s += __shfl_xor(s, off, 32);
  float mean = s * (1.f / 64.f);
  float d0 = x0 - mean, d1 = x1 - mean;
  float v = d0 * d0 + d1 * d1;
#pragma unroll
  for (int off = 16; off > 0; off >>= 1) v += __shfl_xor(v, off, 32);
  float inv = rsqrtf(v * (1.f / 64.f) + 1e-5f);
  out[base + lane]      = d0 * inv * g[lane]      + bt[lane];
  out[base + lane + 32] = d1 * inv * g[lane + 32] + bt[lane + 32];
}

static inline dim3 g1(long long n) { return dim3((unsigned)((n + 255) / 256)); }

extern "C" void kernel_launch(void* const* d_in, const int* in_sizes, int n_in,
                              void* d_out, int out_size, void* d_ws, size_t ws_size,
                              hipStream_t stream) {
  (void)in_sizes; (void)n_in; (void)out_size; (void)ws_size;
  const float* x          = (const float*)d_in[0];
  const float* init_state = (const float*)d_in[1];
  const float* ne         = (const float*)d_in[2];
  const float* te         = (const float*)d_in[3];
  const float* gate_wp    = (const float*)d_in[4];
  const float* gate_wp2   = (const float*)d_in[5];
  const float* gate_bp    = (const float*)d_in[6];
  const float* upd_wp     = (const float*)d_in[7];
  const float* upd_wp2    = (const float*)d_in[8];
  const float* upd_bp     = (const float*)d_in[9];
  const float* vff_w      = (const float*)d_in[10];
  const float* vff_b      = (const float*)d_in[11];
  const float* ln_g       = (const float*)d_in[12];
  const float* ln_b       = (const float*)d_in[13];
  const float* ff_w1      = (const float*)d_in[14];
  const float* ff_b1      = (const float*)d_in[15];
  const float* ff_w2      = (const float*)d_in[16];
  const float* ff_b2      = (const float*)d_in[17];
  const float* lnff_g     = (const float*)d_in[18];
  const float* lnff_b     = (const float*)d_in[19];

  float* ws    = (float*)d_ws;
  float* Ab    = ws + A_OFF;
  float* Adrop = ws + AD_OFF;
  float* biasG = ws + BG_OFF;
  float* biasU = ws + BU_OFF;
  float* state = ws + ST_OFF;
  float* seq   = ws + SEQ_OFF;
  float* Sb    = ws + S_OFF;
  float* ins   = ws + INS_OFF;
  float* cand  = ws + CAND_OFF;
  float* Ax    = ws + AX_OFF;
  float* xg    = ws + XG_OFF;
  float* h0g   = ws + H0G_OFF;
  float* zr    = ws + ZR_OFF;
  float* h0u   = ws + H0U_OFF;
  float* hu    = ws + HU_OFF;
  float* V1    = ws + V1_OFF;
  float* V2    = ws + V2_OFF;
  float* V3    = ws + V3_OFF;

  // ---- static precompute ----
  adj_kernel<<<N_, 256, 0, stream>>>(ne, Ab);
  bias_kernel<<<g1((long long)N_ * H2), 256, 0, stream>>>(ne, gate_bp, biasG, H2);
  bias_kernel<<<g1((long long)N_ * H_), 256, 0, stream>>>(ne, upd_bp, biasU, H_);
  timeS_kernel<<<1, 32, 0, stream>>>(te, Sb);
  copy_kernel<<<g1((long long)B_ * N_ * H_), 256, 0, stream>>>(state, init_state, (long long)B_ * N_ * H_);

  const long long NN = (long long)N_ * N_;
  const dim3 gg65(32, 2, 8), gg128(32, 2, 8), gg64(32, 1, 8);   // 64x64 block tiles

  // ---- GRU scan over T ----
  for (int t = 0; t < T_; ++t) {
    // gate GCN
    pack_in_kernel<<<g1((long long)B_ * N_ * CIN), 256, 0, stream>>>(x, t, state, nullptr, ins);
    graph_gemm<<<gg65, 256, 0, stream>>>(Ab, ins, Ax, CIN);
    pack_xg_kernel<<<g1((long long)B_ * N_ * 2 * CIN), 256, 0, stream>>>(ins, Ax, xg, CIN, -1);
    node_gemm<<<N_, 256, 0, stream>>>(xg, gate_wp, ne, biasG, nullptr, h0g, 2 * CIN, CIN, H2, 0);
    drop_adj_kernel<<<g1(NN), 256, 0, stream>>>(Ab, Adrop, 2 * t + 0);
    graph_gemm<<<gg128, 256, 0, stream>>>(Adrop, h0g, Ax, H2);
    pack_xg_kernel<<<g1((long long)B_ * N_ * 2 * H2), 256, 0, stream>>>(h0g, Ax, xg, H2, 2 * t + 0);
    node_gemm<<<N_, 256, 0, stream>>>(xg, gate_wp2, ne, biasG, h0g, zr, 2 * H2, H2, H2, 1); // sigmoid -> zr
    // candidate GCN
    pack_in_kernel<<<g1((long long)B_ * N_ * CIN), 256, 0, stream>>>(x, t, state, zr, cand);
    graph_gemm<<<gg65, 256, 0, stream>>>(Ab, cand, Ax, CIN);
    pack_xg_kernel<<<g1((long long)B_ * N_ * 2 * CIN), 256, 0, stream>>>(cand, Ax, xg, CIN, -1);
    node_gemm<<<N_, 256, 0, stream>>>(xg, upd_wp, ne, biasU, nullptr, h0u, 2 * CIN, CIN, H_, 0);
    drop_adj_kernel<<<g1(NN), 256, 0, stream>>>(Ab, Adrop, 2 * t + 1);
    graph_gemm<<<gg64, 256, 0, stream>>>(Adrop, h0u, Ax, H_);
    pack_xg_kernel<<<g1((long long)B_ * N_ * 2 * H_), 256, 0, stream>>>(h0u, Ax, xg, H_, 2 * t + 1);
    node_gemm<<<N_, 256, 0, stream>>>(xg, upd_wp2, ne, biasU, h0u, hu, 2 * H_, H_, H_, 2);  // tanh -> hc
    gru_update_kernel<<<g1((long long)B_ * N_ * H_), 256, 0, stream>>>(zr, hu, state, seq, t);
  }

  // ---- temporal attention + FF ----
  const long long BTNH = (long long)B_ * T_ * N_ * H_;
  gemm_ff<<<M_FF / 128, 256, 0, stream>>>(seq, vff_w, vff_b, V1, M_FF, 0);       // v = seq@vff_w+b
  time_mix_kernel<<<g1(BTNH), 256, 0, stream>>>(Sb, V1, V2, 0);                  // v = S-mix (== v0)
  time_mix_kernel<<<g1(BTNH), 256, 0, stream>>>(Sb, V2, V1, 1);                  // v = (S.*drop)-mix + v0
  ln_kernel<<<(M_FF + 7) / 8, 256, 0, stream>>>(V1, seq, ln_g, ln_b, V2, M_FF);  // v = LN(v + seq)
  gemm_ff<<<M_FF / 128, 256, 0, stream>>>(V2, ff_w1, ff_b1, V3, M_FF, 1);        // relu(v@W1+b1)
  gemm_ff<<<M_FF / 128, 256, 0, stream>>>(V3, ff_w2, ff_b2, V1, M_FF, 0);        // @W2+b2
  ln_kernel<<<(M_FF + 7) / 8, 256, 0, stream>>>(V1, V2, lnff_g, lnff_b, (float*)d_out, M_FF); // out = LN(ff+v)
  copy_kernel<<<g1((long long)B_ * N_ * H_), 256, 0, stream>>>((float*)d_out + SEQ_SZ, state,
                                                               (long long)B_ * N_ * H_);      // last_state
}